// QCNN_ZNOTY_OLD_12756052869699
// MI455X (gfx1250) — compile-verified
//
#include <hip/hip_runtime.h>
#include <math.h>

// ---------------------------------------------------------------------------
// QCNN forward for MI455X (gfx1250):
//   Kernel 1 (tiny): simulate the batch-independent circuit on 256 basis
//                    states -> 256x256 complex unitary U in workspace.
//   Kernel 2 (main): Phi = Psi * U^T as f32 complex GEMM on
//                    v_wmma_f32_16x16x4_f32, with U K-chunks double-buffered
//                    into LDS via global_load_async_to_lds_b128 (ASYNCcnt),
//                    then Z-weighted |Phi|^2 reduction + MLP + sigmoid.
// ---------------------------------------------------------------------------

typedef __attribute__((ext_vector_type(2))) float v2f;
typedef __attribute__((ext_vector_type(8))) float v8f;
typedef int qv4i __attribute__((vector_size(16)));  // builtin's pointee type

#define NQ 8
#define NSTATE 256
#define BATCH 65536
#define PI_F 3.14159265358979323846f

// ---------------- async-copy plumbing (gfx1250) ----------------------------
#if defined(__has_builtin)
#if __has_builtin(__builtin_amdgcn_global_load_async_to_lds_b128)
#define QCNN_ASYNC_BUILTIN 1
#endif
#if __has_builtin(__builtin_amdgcn_s_wait_asynccnt)
#define QCNN_WAIT_BUILTIN 1
#endif
#endif

__device__ __forceinline__ void async_copy16(const float* g, float* l) {
#ifdef QCNN_ASYNC_BUILTIN
  __builtin_amdgcn_global_load_async_to_lds_b128(
      (__attribute__((address_space(1))) qv4i*)g,
      (__attribute__((address_space(3))) qv4i*)l, 0, 0);
#else
  unsigned lofs =
      (unsigned)(__UINTPTR_TYPE__)(__attribute__((address_space(3))) float*)l;
  asm volatile("global_load_async_to_lds_b128 %0, %1, off" ::"v"(lofs), "v"(g)
               : "memory");
#endif
}

#ifdef QCNN_WAIT_BUILTIN
#define WAIT_ASYNC(n) __builtin_amdgcn_s_wait_asynccnt(n)
#else
#define WAIT_ASYNC(n) asm volatile("s_wait_asynccnt %0" ::"i"(n) : "memory")
#endif

// ------------------------------ Kernel 1 -----------------------------------
struct C2 { float re, im; };
struct M2 { C2 m00, m01, m10, m11; };

__device__ __forceinline__ M2 rz_m(float t) {
  float c = __cosf(0.5f * t), s = __sinf(0.5f * t);
  M2 u; u.m00 = { c, -s }; u.m01 = { 0.f, 0.f };
  u.m10 = { 0.f, 0.f };    u.m11 = { c,  s };
  return u;
}
__device__ __forceinline__ M2 ry_m(float t) {
  float c = __cosf(0.5f * t), s = __sinf(0.5f * t);
  M2 u; u.m00 = { c, 0.f }; u.m01 = { -s, 0.f };
  u.m10 = { s, 0.f };       u.m11 = { c,  0.f };
  return u;
}
__device__ __forceinline__ M2 u3_m(float t, float p, float l) {
  float c = __cosf(0.5f * t), s = __sinf(0.5f * t);
  float cp = __cosf(p), sp = __sinf(p);
  float cl = __cosf(l), sl = __sinf(l);
  float cpl = __cosf(p + l), spl = __sinf(p + l);
  M2 u;
  u.m00 = { c, 0.f };
  u.m01 = { -cl * s, -sl * s };
  u.m10 = { cp * s,  sp * s };
  u.m11 = { cpl * c, spl * c };
  return u;
}

// wire w maps to bit position (7 - w) of the flat amplitude index
__device__ void apply1q(float* sre, float* sim, int w, M2 u) {
  const int pos = 7 - w;
  const int mask = 1 << pos;
  for (int p = threadIdx.x; p < 128; p += 32) {
    int i0 = ((p >> pos) << (pos + 1)) | (p & (mask - 1));
    int i1 = i0 | mask;
    float r0 = sre[i0], q0 = sim[i0];
    float r1 = sre[i1], q1 = sim[i1];
    sre[i0] = u.m00.re * r0 - u.m00.im * q0 + u.m01.re * r1 - u.m01.im * q1;
    sim[i0] = u.m00.re * q0 + u.m00.im * r0 + u.m01.re * q1 + u.m01.im * r1;
    sre[i1] = u.m10.re * r0 - u.m10.im * q0 + u.m11.re * r1 - u.m11.im * q1;
    sim[i1] = u.m10.re * q0 + u.m10.im * r0 + u.m11.re * q1 + u.m11.im * r1;
  }
  __syncthreads();
}

__device__ void applyCNOT(float* sre, float* sim, int cw, int tw) {
  const int cpos = 7 - cw, tpos = 7 - tw;
  const int p1 = cpos < tpos ? cpos : tpos;
  const int p2 = cpos < tpos ? tpos : cpos;
  for (int q = threadIdx.x; q < 64; q += 32) {
    int t1 = ((q  >> p1) << (p1 + 1)) | (q  & ((1 << p1) - 1));
    int t2 = ((t1 >> p2) << (p2 + 1)) | (t1 & ((1 << p2) - 1));
    int i0 = t2 | (1 << cpos);
    int i1 = i0 | (1 << tpos);
    float tr = sre[i0]; sre[i0] = sre[i1]; sre[i1] = tr;
    float ti = sim[i0]; sim[i0] = sim[i1]; sim[i1] = ti;
  }
  __syncthreads();
}

__device__ void do_block(float* sre, float* sim, int a, int b,
                         float rzv, float ry1, float ry2, int fin) {
  apply1q(sre, sim, a, rz_m(-PI_F * 0.5f));
  applyCNOT(sre, sim, a, b);
  apply1q(sre, sim, b, rz_m(rzv));
  apply1q(sre, sim, a, ry_m(ry1));
  applyCNOT(sre, sim, b, a);
  apply1q(sre, sim, a, ry_m(ry2));
  applyCNOT(sre, sim, a, b);
  apply1q(sre, sim, fin, rz_m(PI_F * 0.5f));
}

__global__ __launch_bounds__(32) void build_U_kernel(
    const float* __restrict__ rzp, const float* __restrict__ ryp,
    const float* __restrict__ u3p, float* __restrict__ Ure,
    float* __restrict__ Uim) {
  __shared__ float sre[NSTATE];
  __shared__ float sim[NSTATE];
  const int col = blockIdx.x;
  for (int i = threadIdx.x; i < NSTATE; i += 32) {
    sre[i] = (i == col) ? 1.f : 0.f;
    sim[i] = 0.f;
  }
  __syncthreads();

  do_block(sre, sim, 1, 0, rzp[1],  ryp[0],  ryp[1],  0);
  do_block(sre, sim, 3, 2, rzp[4],  ryp[2],  ryp[3],  2);
  do_block(sre, sim, 5, 4, rzp[7],  ryp[4],  ryp[5],  4);
  do_block(sre, sim, 7, 6, rzp[10], ryp[6],  ryp[7],  6);
  apply1q(sre, sim, 1, u3_m(u3p[0], u3p[1],  u3p[2]));
  apply1q(sre, sim, 3, u3_m(u3p[3], u3p[4],  u3p[5]));
  apply1q(sre, sim, 5, u3_m(u3p[6], u3p[7],  u3p[8]));
  apply1q(sre, sim, 7, u3_m(u3p[9], u3p[10], u3p[11]));
  do_block(sre, sim, 3, 1, rzp[13], ryp[8],  ryp[9],  1);
  do_block(sre, sim, 7, 5, rzp[16], ryp[10], ryp[11], 1);
  do_block(sre, sim, 5, 3, rzp[19], ryp[12], ryp[13], 3);
  apply1q(sre, sim, 3, u3_m(u3p[12], u3p[13], u3p[14]));
  apply1q(sre, sim, 7, u3_m(u3p[15], u3p[16], u3p[17]));

  // U[:, col]; store row-major U[n][m] so the GEMM reads contiguous K
  for (int i = threadIdx.x; i < NSTATE; i += 32) {
    Ure[i * NSTATE + col] = sre[i];
    Uim[i * NSTATE + col] = sim[i];
  }
}

// ------------------------------ Kernel 2 -----------------------------------
// 16 batch rows per block; 4 waves; each wave owns 4 N-tiles of 16 columns.
// K = 256 processed in 16 chunks of 16; each chunk's U slice is staged into
// LDS with per-wave async b128 copies, double-buffered against WMMA compute.
#define KCH 16        // K elements per chunk
#define NCHUNK (NSTATE / KCH)
#define BPAD 20       // padded floats per staged row (80B: 16B-aligned, bank-spread)

__global__ __launch_bounds__(128) void qcnn_gemm_kernel(
    const float* __restrict__ x, const float* __restrict__ Ure,
    const float* __restrict__ Uim, const float* __restrict__ W1,
    const float* __restrict__ b1, const float* __restrict__ W2,
    const float* __restrict__ b2, float* __restrict__ out) {
  __shared__ float PsiRe[16][NSTATE];
  __shared__ float PsiIm[16][NSTATE];
  __shared__ float Bbuf[2][16][2][16][BPAD];  // [buf][ntile][re/im][n][k] ~80KB
  __shared__ float tpw[16][9], tco[16][9], tsi[16][9];
  __shared__ float feat[16][2];

  const int tid = threadIdx.x;
  const int b0 = blockIdx.x * 16;

  if (tid < 32) feat[tid >> 1][tid & 1] = 0.f;

  // Closed-form per-row tables: |psi_m| = c^(8-k) s^k, phase = (k-4)*phi
  for (int i = tid; i < 16 * 9; i += 128) {
    int r = i / 9, k = i % 9;
    float th = x[(b0 + r) * 2 + 0];
    float ph = x[(b0 + r) * 2 + 1];
    float s, c;
    __sincosf(0.5f * th, &s, &c);
    float pw = 1.f;
    for (int j = 0; j < 8 - k; ++j) pw *= c;
    for (int j = 0; j < k; ++j) pw *= s;
    float sv, cv;
    __sincosf((float)(k - 4) * ph, &sv, &cv);
    tpw[r][k] = pw; tco[r][k] = cv; tsi[r][k] = sv;
  }
  __syncthreads();

  // Materialize Psi tile (16 x 256 complex) in LDS
  for (int i = tid; i < 16 * NSTATE; i += 128) {
    int r = i >> 8, m = i & 255;
    int k = __popc(m);
    float pw = tpw[r][k];
    PsiRe[r][m] = pw * tco[r][k];
    PsiIm[r][m] = pw * tsi[r][k];
  }
  __syncthreads();

  const int wave = tid >> 5;
  const int lane = tid & 31;
  const int Ml = lane & 15;       // A-matrix M row / B-matrix N column per lane
  const bool hiHalf = lane >= 16; // lanes 16..31 carry K = 2,3 of each K-step

  // Stage one K-chunk of this wave's 4 N-tiles (re+im) into LDS buffer `buf`.
  // 16 async b128 issues per wave per chunk, tracked by ASYNCcnt only.
  auto stage = [&](int c, int buf) {
#pragma unroll
    for (int nti = 0; nti < 4; ++nti) {
      const int nt = wave * 4 + nti;
#pragma unroll
      for (int plane = 0; plane < 2; ++plane) {
        const float* base = plane ? Uim : Ure;
#pragma unroll
        for (int j = 0; j < 2; ++j) {
          int slot = j * 32 + lane;       // 64 slots = 16 rows * 4 fragments
          int r = slot >> 2;
          int f = slot & 3;
          const float* g = base + (nt * 16 + r) * NSTATE + c * KCH + f * 4;
          float* l = &Bbuf[buf][nt][plane][r][f * 4];
          async_copy16(g, l);
        }
      }
    }
  };

  v8f cre[4], cim[4];
#pragma unroll
  for (int nti = 0; nti < 4; ++nti) { cre[nti] = (v8f){}; cim[nti] = (v8f){}; }

  stage(0, 0);
  for (int c = 0; c < NCHUNK; ++c) {
    if (c + 1 < NCHUNK) {
      stage(c + 1, (c + 1) & 1);
      WAIT_ASYNC(16);   // chunk c complete; chunk c+1 still in flight
    } else {
      WAIT_ASYNC(0);
    }
    asm volatile("" ::: "memory");
    const int buf = c & 1;

#pragma unroll
    for (int nti = 0; nti < 4; ++nti) {
      const int nt = wave * 4 + nti;
#pragma unroll
      for (int ks = 0; ks < 4; ++ks) {
        const int m0l = ks * 4 + (hiHalf ? 2 : 0);
        const int m0 = c * KCH + m0l;
        v2f are = { PsiRe[Ml][m0], PsiRe[Ml][m0 + 1] };
        v2f aim = { PsiIm[Ml][m0], PsiIm[Ml][m0 + 1] };
        v2f ain = { -aim.x, -aim.y };
        v2f bre = *(const v2f*)&Bbuf[buf][nt][0][Ml][m0l];
        v2f bim = *(const v2f*)&Bbuf[buf][nt][1][Ml][m0l];
        // complex MAC as 4 real WMMA contributions
        cre[nti] = __builtin_amdgcn_wmma_f32_16x16x4_f32(
            false, are, false, bre, (short)0, cre[nti], false, false);
        cre[nti] = __builtin_amdgcn_wmma_f32_16x16x4_f32(
            false, ain, false, bim, (short)0, cre[nti], false, false);
        cim[nti] = __builtin_amdgcn_wmma_f32_16x16x4_f32(
            false, are, false, bim, (short)0, cim[nti], false, false);
        cim[nti] = __builtin_amdgcn_wmma_f32_16x16x4_f32(
            false, aim, false, bre, (short)0, cim[nti], false, false);
      }
    }
  }

  // Z-weighted |Phi|^2: wire3 -> bit4 of n (constant per N-tile), wire7 -> bit0
  float f0acc[8], f1acc[8];
#pragma unroll
  for (int r = 0; r < 8; ++r) { f0acc[r] = 0.f; f1acc[r] = 0.f; }
#pragma unroll
  for (int nti = 0; nti < 4; ++nti) {
    const int nt = wave * 4 + nti;
    const int n = nt * 16 + Ml;
    const float z3 = (nt & 1) ? -1.f : 1.f;
    const float z7 = (n & 1) ? -1.f : 1.f;
#pragma unroll
    for (int r = 0; r < 8; ++r) {
      float v = cre[nti][r] * cre[nti][r] + cim[nti][r] * cim[nti][r];
      f0acc[r] += z3 * v;
      f1acc[r] += z7 * v;
    }
  }

  // Reduce over the 16 lanes sharing each M row (xor masks stay in-half)
#pragma unroll
  for (int msk = 1; msk < 16; msk <<= 1) {
#pragma unroll
    for (int r = 0; r < 8; ++r) {
      f0acc[r] += __shfl_xor(f0acc[r], msk, 32);
      f1acc[r] += __shfl_xor(f1acc[r], msk, 32);
    }
  }
  if (Ml == 0) {
#pragma unroll
    for (int r = 0; r < 8; ++r) {
      int M = r + (hiHalf ? 8 : 0);
      atomicAdd(&feat[M][0], f0acc[r]);   // ds_add_f32
      atomicAdd(&feat[M][1], f1acc[r]);
    }
  }
  __syncthreads();

  // Tiny MLP head: tanh(feats @ W1 + b1) @ W2 + b2 -> sigmoid
  if (tid < 16) {
    float f0 = feat[tid][0], f1 = feat[tid][1];
    float acc = b2[0];
#pragma unroll
    for (int j = 0; j < 10; ++j) {
      float h = tanhf(f0 * W1[j] + f1 * W1[10 + j] + b1[j]);
      acc += h * W2[j];
    }
    out[b0 + tid] = 1.f / (1.f + __expf(-acc));
  }
}

// ------------------------------ Launcher -----------------------------------
extern "C" void kernel_launch(void* const* d_in, const int* in_sizes, int n_in,
                              void* d_out, int out_size, void* d_ws,
                              size_t ws_size, hipStream_t stream) {
  const float* x   = (const float*)d_in[0];
  const float* rzp = (const float*)d_in[1];
  const float* ryp = (const float*)d_in[2];
  const float* u3p = (const float*)d_in[3];
  const float* W1  = (const float*)d_in[4];
  const float* b1  = (const float*)d_in[5];
  const float* W2  = (const float*)d_in[6];
  const float* b2  = (const float*)d_in[7];

  float* Ure = (float*)d_ws;                 // 256*256 f32 = 256 KB
  float* Uim = Ure + NSTATE * NSTATE;        // +256 KB

  build_U_kernel<<<NSTATE, 32, 0, stream>>>(rzp, ryp, u3p, Ure, Uim);
  qcnn_gemm_kernel<<<BATCH / 16, 128, 0, stream>>>(x, Ure, Uim, W1, b1, W2, b2,
                                                   (float*)d_out);
}